// GaussianCloud_14929306321119
// MI455X (gfx1250) — compile-verified
//
#include <hip/hip_runtime.h>

#define TILE 256

// ---- CDNA5 async global->LDS path (probe via __has_builtin) --------------
#if defined(__has_builtin)
#if __has_builtin(__builtin_amdgcn_global_load_async_to_lds_b128) && \
    __has_builtin(__builtin_amdgcn_s_wait_asynccnt)
#define USE_ASYNC_LDS 1
#endif
#endif
#ifndef USE_ASYNC_LDS
#define USE_ASYNC_LDS 0
#endif

#define AS1 __attribute__((address_space(1)))
#define AS3 __attribute__((address_space(3)))

typedef int v4i __attribute__((vector_size(16)));
typedef float v4f __attribute__((vector_size(16)));  // native clang vector (16B)
typedef AS1 v4i* async_gptr_t;  // global source ("V4i __device__ *")
typedef AS3 v4i* async_lptr_t;  // LDS destination

__device__ __forceinline__ void gauss_cov_math(const float s_raw[3], const float q[4],
                                               float cov[9]) {
    const float e0 = expf(s_raw[0]);
    const float e1 = expf(s_raw[1]);
    const float e2 = expf(s_raw[2]);

    const float nrm = sqrtf(q[0] * q[0] + q[1] * q[1] + q[2] * q[2] + q[3] * q[3]);
    const float inv = 1.0f / fmaxf(nrm, 1e-12f);
    const float w = q[0] * inv, x = q[1] * inv, y = q[2] * inv, z = q[3] * inv;

    const float r00 = 1.0f - 2.0f * (y * y + z * z);
    const float r01 = 2.0f * (x * y - w * z);
    const float r02 = 2.0f * (x * z + w * y);
    const float r10 = 2.0f * (x * y + w * z);
    const float r11 = 1.0f - 2.0f * (x * x + z * z);
    const float r12 = 2.0f * (y * z - w * x);
    const float r20 = 2.0f * (x * z - w * y);
    const float r21 = 2.0f * (y * z + w * x);
    const float r22 = 1.0f - 2.0f * (x * x + y * y);

    // RS = R * diag(exp(s)) : scale columns
    const float a0 = r00 * e0, a1 = r01 * e1, a2 = r02 * e2;
    const float b0 = r10 * e0, b1 = r11 * e1, b2 = r12 * e2;
    const float c0 = r20 * e0, c1 = r21 * e1, c2 = r22 * e2;

    // cov = RS * RS^T (symmetric)
    cov[0] = a0 * a0 + a1 * a1 + a2 * a2;
    cov[1] = a0 * b0 + a1 * b1 + a2 * b2;
    cov[2] = a0 * c0 + a1 * c1 + a2 * c2;
    cov[3] = cov[1];
    cov[4] = b0 * b0 + b1 * b1 + b2 * b2;
    cov[5] = b0 * c0 + b1 * c1 + b2 * c2;
    cov[6] = cov[2];
    cov[7] = cov[5];
    cov[8] = c0 * c0 + c1 * c1 + c2 * c2;
}

__global__ __launch_bounds__(TILE) void gauss_cov_kernel(const float* __restrict__ scales,
                                                         const float* __restrict__ rots,
                                                         float* __restrict__ out, int n) {
    // Reused staging buffer: inputs need 1792 floats, outputs need 2304 floats.
    __shared__ float lds[TILE * 9];

    const int tid = threadIdx.x;
    const long long base = (long long)blockIdx.x * TILE;
    const bool full = (base + TILE) <= (long long)n;

    float s[3];
    float q[4];

    if (full) {
        // ---- stage inputs into LDS, coalesced 16B per lane, async data mover ----
        const v4f* sg4 = (const v4f*)(scales + base * 3);  // 192 x 16B
        const v4f* rg4 = (const v4f*)(rots + base * 4);    // 256 x 16B
        v4f* l4 = (v4f*)lds;
#if USE_ASYNC_LDS
        if (tid < 192) {
            __builtin_amdgcn_global_load_async_to_lds_b128(
                (async_gptr_t)(sg4 + tid), (async_lptr_t)(l4 + tid), 0, 0);
        }
        __builtin_amdgcn_global_load_async_to_lds_b128(
            (async_gptr_t)(rg4 + tid), (async_lptr_t)(l4 + 192 + tid), 0, 0);
        __builtin_amdgcn_s_wait_asynccnt(0);
#else
        if (tid < 192) l4[tid] = sg4[tid];
        l4[192 + tid] = rg4[tid];
#endif
        __syncthreads();

        // stride-3 read: conflict-free across 64 banks
        s[0] = lds[tid * 3 + 0];
        s[1] = lds[tid * 3 + 1];
        s[2] = lds[tid * 3 + 2];
        // stride-4 read
        q[0] = lds[768 + tid * 4 + 0];
        q[1] = lds[768 + tid * 4 + 1];
        q[2] = lds[768 + tid * 4 + 2];
        q[3] = lds[768 + tid * 4 + 3];

        __syncthreads();  // all input reads done before the buffer is reused
    } else {
        const long long idx = base + tid;
        if (idx < n) {
            s[0] = scales[idx * 3 + 0];
            s[1] = scales[idx * 3 + 1];
            s[2] = scales[idx * 3 + 2];
            q[0] = rots[idx * 4 + 0];
            q[1] = rots[idx * 4 + 1];
            q[2] = rots[idx * 4 + 2];
            q[3] = rots[idx * 4 + 3];
        } else {
            s[0] = s[1] = s[2] = 0.0f;
            q[0] = 1.0f; q[1] = q[2] = q[3] = 0.0f;
        }
    }

    float cov[9];
    gauss_cov_math(s, q, cov);

    if (full) {
        // stride-9 write: conflict-free across 64 banks
#pragma unroll
        for (int k = 0; k < 9; ++k) lds[tid * 9 + k] = cov[k];
        __syncthreads();

        // ---- coalesced non-temporal b128 stores: 576 x 16B per block ----
        // Output is write-once / never re-read within the kernel: NT keeps the
        // stream from displacing L2 lines (256 MB stream > 192 MB L2 anyway).
        v4f* og4 = (v4f*)(out + base * 9);  // base*36 bytes: 16B aligned
        const v4f* l4o = (const v4f*)lds;
        __builtin_nontemporal_store(l4o[tid], &og4[tid]);
        __builtin_nontemporal_store(l4o[tid + 256], &og4[tid + 256]);
        if (tid < 64) __builtin_nontemporal_store(l4o[tid + 512], &og4[tid + 512]);
    } else {
        const long long idx = base + tid;
        if (idx < n) {
#pragma unroll
            for (int k = 0; k < 9; ++k) out[idx * 9 + k] = cov[k];
        }
    }
}

extern "C" void kernel_launch(void* const* d_in, const int* in_sizes, int n_in,
                              void* d_out, int out_size, void* d_ws, size_t ws_size,
                              hipStream_t stream) {
    (void)n_in; (void)d_ws; (void)ws_size; (void)out_size;
    const float* scales = (const float*)d_in[0];  // (N,3) f32
    const float* rots   = (const float*)d_in[1];  // (N,4) f32
    float* out = (float*)d_out;                   // (N,3,3) f32

    const int n = in_sizes[0] / 3;
    const int grid = (n + TILE - 1) / TILE;
    gauss_cov_kernel<<<grid, TILE, 0, stream>>>(scales, rots, out, n);
}